// LearnedPositionalBias_8074538516950
// MI455X (gfx1250) — compile-verified
//
#include <hip/hip_runtime.h>
#include <hip/hip_bf16.h>

typedef __attribute__((ext_vector_type(2))) float v2f;
typedef __attribute__((ext_vector_type(8))) float v8f;

#define N_BATCH 4
#define H_HEADS 16
#define L_LEN   1024
#define D_DIM   64
#define J_LEN   1025   // 2*512+1
#define MAXPOS  512
#define ROWS_PER_BLOCK 32      // two 16-row strips share every B tile
#define LDS_BYTES (ROWS_PER_BLOCK * L_LEN * 4)   // 128 KB dynamic LDS

// One workgroup = one (n,h) and 32 rows (two 16-row strips). 8 waves.
// Phase 1: wave w owns table columns [w*128, w*128+128). For each 16-wide
//   column tile it loads the B fragment (rel_emb chunk) ONCE and feeds it to
//   TWO V_WMMA_F32_16X16X4_F32 chains (one per row-strip), halving rel_emb
//   L2 traffic vs one strip per block. K=64 -> 16 chained WMMAs per acc.
//   Results go to a 32x1024 f32 LDS strip (128 KB dynamic LDS; WGP has 320KB).
// Phase 2: gather with t = clip(j-i,-511,511)+512 (always in [1,1023]) from
//   LDS and write coalesced 16B float4 stores (the HBM-bound part: 256 MB
//   total at 23.3 TB/s ~ 11 us is this kernel's floor).
__global__ __launch_bounds__(256) void lpb_fused_kernel(
    const float* __restrict__ q,     // [n, h, L, d]
    const float* __restrict__ rel,   // [h, d, 2*maxpos+1]
    float* __restrict__ out) {       // [n, h, L, L]
  extern __shared__ float smem[];    // [32][1024] logits strip

  const int wg  = blockIdx.x;
  const int nh  = wg >> 5;                      // 32 blocks per (n,h)
  const int i0  = (wg & 31) * ROWS_PER_BLOCK;   // first row of this block
  const int h   = nh & (H_HEADS - 1);

  const int lane  = threadIdx.x & 31;
  const int wave  = threadIdx.x >> 5;
  const int row   = lane & 15;             // M for A, N for B/C/D
  const int khalf = lane >> 4;             // selects K pair {0,1} vs {2,3}

  // ---- Phase 1: WMMA GEMM into LDS strips ------------------------------
  const float* qrow0 = q + (size_t)nh * (L_LEN * D_DIM)
                         + (size_t)(i0 + row) * D_DIM;
  const float* qrow1 = qrow0 + 16 * D_DIM;
  const float* relh  = rel + (size_t)h * (D_DIM * J_LEN);

  // A fragments for both strips, all 16 K-chunks (K=4 each): 64 VGPRs.
  v2f a0[16], a1[16];
#pragma unroll
  for (int kc = 0; kc < 16; ++kc) {
    // lane<16: {K=4kc,4kc+1}; lane>=16: {K=4kc+2,4kc+3}  (8B aligned)
    a0[kc] = *reinterpret_cast<const v2f*>(qrow0 + 4 * kc + 2 * khalf);
    a1[kc] = *reinterpret_cast<const v2f*>(qrow1 + 4 * kc + 2 * khalf);
  }

#pragma unroll
  for (int tile = 0; tile < 8; ++tile) {
    const int t0 = wave * 128 + tile * 16;   // column tile of the table
    v8f acc0 = {};
    v8f acc1 = {};
#pragma unroll
    for (int kc = 0; kc < 16; ++kc) {
      const int d0 = 4 * kc + 2 * khalf;     // B rows K=d0, d0+1 this half
      const float* rp = relh + (size_t)d0 * J_LEN + (t0 + row);
      v2f b;
      b.x = rp[0];
      b.y = rp[J_LEN];
      // B loaded once, used by both strips.
      acc0 = __builtin_amdgcn_wmma_f32_16x16x4_f32(
          false, a0[kc], false, b, (short)0, acc0, false, false);
      acc1 = __builtin_amdgcn_wmma_f32_16x16x4_f32(
          false, a1[kc], false, b, (short)0, acc1, false, false);
    }
    // D layout: VGPR v holds (M = v + 8*khalf, N = row)
#pragma unroll
    for (int v = 0; v < 8; ++v) {
      smem[(size_t)(v + 8 * khalf) * L_LEN + (t0 + row)]      = acc0[v];
      smem[(size_t)(16 + v + 8 * khalf) * L_LEN + (t0 + row)] = acc1[v];
    }
  }

  __syncthreads();

  // ---- Phase 2: clamped diagonal gather + coalesced float4 stores ------
  float* ostrip = out + (size_t)nh * ((size_t)L_LEN * L_LEN)
                      + (size_t)i0 * L_LEN;
  const int j = (int)threadIdx.x * 4;        // 256 threads * 4 = 1024 cols

#pragma unroll 4
  for (int r = 0; r < ROWS_PER_BLOCK; ++r) {
    const int i = i0 + r;
    const float* srow = smem + (size_t)r * L_LEN;
    float4 vres;
    {
      int k = j + 0 - i;
      k = k < -(MAXPOS - 1) ? -(MAXPOS - 1) : (k > (MAXPOS - 1) ? (MAXPOS - 1) : k);
      vres.x = srow[k + MAXPOS];
    }
    {
      int k = j + 1 - i;
      k = k < -(MAXPOS - 1) ? -(MAXPOS - 1) : (k > (MAXPOS - 1) ? (MAXPOS - 1) : k);
      vres.y = srow[k + MAXPOS];
    }
    {
      int k = j + 2 - i;
      k = k < -(MAXPOS - 1) ? -(MAXPOS - 1) : (k > (MAXPOS - 1) ? (MAXPOS - 1) : k);
      vres.z = srow[k + MAXPOS];
    }
    {
      int k = j + 3 - i;
      k = k < -(MAXPOS - 1) ? -(MAXPOS - 1) : (k > (MAXPOS - 1) ? (MAXPOS - 1) : k);
      vres.w = srow[k + MAXPOS];
    }
    *reinterpret_cast<float4*>(ostrip + (size_t)r * L_LEN + j) = vres;
  }
}

extern "C" void kernel_launch(void* const* d_in, const int* in_sizes, int n_in,
                              void* d_out, int out_size, void* d_ws, size_t ws_size,
                              hipStream_t stream) {
  (void)in_sizes; (void)n_in; (void)out_size; (void)d_ws; (void)ws_size;
  const float* q   = (const float*)d_in[0];   // [4,16,1024,64]
  const float* rel = (const float*)d_in[1];   // [16,64,1025]
  float* out = (float*)d_out;                 // [4,16,1024,1024]

  const int blocks_per_nh = L_LEN / ROWS_PER_BLOCK;        // 32
  const int grid = N_BATCH * H_HEADS * blocks_per_nh;      // 2048 workgroups
  lpb_fused_kernel<<<dim3(grid), dim3(256), LDS_BYTES, stream>>>(q, rel, out);
}